// GCNLayer_32229434589218
// MI455X (gfx1250) — compile-verified
//
#include <hip/hip_runtime.h>

// ---------------------------------------------------------------------------
// GCN layer for MI455X (gfx1250, wave32):
//   out = D^{-1/2} A D^{-1/2} (x W + b)
// Inputs (in order): rows[i32 E], cols[i32 E], adj_vals[f32 E],
//                    x[f32 N*64], W[f32 64*64], b[f32 64]
// Output: out[f32 N*64]
// ---------------------------------------------------------------------------

typedef __attribute__((ext_vector_type(2))) float v2f;
typedef __attribute__((ext_vector_type(8))) float v8f;

#define IN_DIM 64
#define OUT_DIM 64

// ---------------------------------------------------------------------------
// Zero-fill (harness poisons d_out / d_ws with 0xAA).
// ---------------------------------------------------------------------------
__global__ __launch_bounds__(256) void zero_f32_kernel(float* __restrict__ p, long long n) {
    long long i = (long long)blockIdx.x * blockDim.x + threadIdx.x;
    if (i < n) p[i] = 0.0f;
}

// ---------------------------------------------------------------------------
// deg[r] += adj_vals[e]  (hardware f32 atomic, no return -> STOREcnt path)
// ---------------------------------------------------------------------------
__global__ __launch_bounds__(256) void degree_kernel(const int* __restrict__ rows,
                                                     const float* __restrict__ adj,
                                                     float* __restrict__ deg,
                                                     int n_edges) {
    int e = blockIdx.x * blockDim.x + threadIdx.x;
    if (e < n_edges) {
        unsafeAtomicAdd(&deg[rows[e]], adj[e]);
    }
}

// ---------------------------------------------------------------------------
// dinv[i] = (deg[i] > 0 ? deg[i] : 1)^{-1/2}
// ---------------------------------------------------------------------------
__global__ __launch_bounds__(256) void dinv_kernel(const float* __restrict__ deg,
                                                   float* __restrict__ dinv,
                                                   int n_nodes) {
    int i = blockIdx.x * blockDim.x + threadIdx.x;
    if (i < n_nodes) {
        float d = deg[i];
        d = (d > 0.0f) ? d : 1.0f;
        dinv[i] = rsqrtf(d);
    }
}

// ---------------------------------------------------------------------------
// Dense GEMM via V_WMMA_F32_16X16X4_F32:  support = x @ W + b
// One wave (32 lanes) computes a 16-row x 64-col block of `support`:
//   4 N-tiles of 16x16, K = 64 in steps of 4.
//
// A-fragment (16x4 f32, ISA 7.12.2): lanes 0-15 -> rows M=0..15, K={k,k+1};
// lanes 16-31 -> same rows, K={k+2,k+3}.  -> contiguous float2 per lane.
// B-fragment (4x16): mirrored layout, row K = k + 2*half + j, col N = n0 + r.
// C/D (16x16 f32, 8 VGPRs): VGPR i -> M=i (lanes 0-15, N=lane) or
// M=i+8 (lanes 16-31, N=lane-16).
// ---------------------------------------------------------------------------
__global__ __launch_bounds__(64) void gcn_gemm_wmma(const float* __restrict__ x,
                                                    const float* __restrict__ W,
                                                    const float* __restrict__ b,
                                                    float* __restrict__ support,
                                                    int n_nodes) {
    const int wave = threadIdx.x >> 5;                 // 0..1 (wave32)
    const int lane = threadIdx.x & 31;
    const int mtile = blockIdx.x * 2 + wave;           // 16-row block index
    const int m0 = mtile * 16;
    if (m0 >= n_nodes) return;                         // wave-uniform: EXEC stays all-1s

    const int half = lane >> 4;                        // 0: lanes 0-15, 1: lanes 16-31
    const int r = lane & 15;

    // Accumulators for the 4 N-tiles, initialized with the bias: C[m][n] = b[n].
    v8f acc[4];
#pragma unroll
    for (int nt = 0; nt < 4; ++nt) {
        const float bv = b[nt * 16 + r];
#pragma unroll
        for (int i = 0; i < 8; ++i) acc[nt][i] = bv;
    }

    const float* arow = x + (size_t)(m0 + r) * IN_DIM;

#pragma unroll
    for (int k = 0; k < IN_DIM; k += 4) {
        // A fragment: contiguous float2 per lane.
        const v2f a = *(const v2f*)(arow + k + 2 * half);
#pragma unroll
        for (int nt = 0; nt < 4; ++nt) {
            v2f bf;
            bf.x = W[(size_t)(k + 2 * half + 0) * OUT_DIM + nt * 16 + r];
            bf.y = W[(size_t)(k + 2 * half + 1) * OUT_DIM + nt * 16 + r];
            // 8 args: (neg_a, A, neg_b, B, c_mod, C, reuse_a, reuse_b)
            acc[nt] = __builtin_amdgcn_wmma_f32_16x16x4_f32(
                false, a, false, bf, (short)0, acc[nt], false, false);
        }
    }

    // Store D per the 16x16 f32 C/D layout.
    float* srow = support + (size_t)m0 * OUT_DIM;
#pragma unroll
    for (int nt = 0; nt < 4; ++nt) {
#pragma unroll
        for (int i = 0; i < 8; ++i) {
            srow[(size_t)(i + 8 * half) * OUT_DIM + nt * 16 + r] = acc[nt][i];
        }
    }
}

// ---------------------------------------------------------------------------
// SpMM scatter: out[rows[e], f] += adj[e]*dinv[rows[e]]*dinv[cols[e]] *
//                                  support[cols[e], f]
// 64 consecutive threads (2 waves) handle one edge's 64 features:
// support reads coalesce (128B per wave), atomics hit 32 consecutive floats
// (L2-resident: support+out fit in the 192MB L2).
// ---------------------------------------------------------------------------
__global__ __launch_bounds__(256) void spmm_scatter_kernel(const int* __restrict__ rows,
                                                           const int* __restrict__ cols,
                                                           const float* __restrict__ adj,
                                                           const float* __restrict__ dinv,
                                                           const float* __restrict__ support,
                                                           float* __restrict__ out,
                                                           int n_edges) {
    long long t = (long long)blockIdx.x * blockDim.x + threadIdx.x;
    int e = (int)(t >> 6);
    int f = (int)(t & 63);
    if (e >= n_edges) return;
    const int rrow = rows[e];
    const int ccol = cols[e];
    const float norm = adj[e] * dinv[rrow] * dinv[ccol];
    const float v = norm * support[(size_t)ccol * OUT_DIM + f];
    unsafeAtomicAdd(&out[(size_t)rrow * OUT_DIM + f], v);
}

// ---------------------------------------------------------------------------
// Host-side launcher (all launches on `stream`, graph-capture safe).
// ---------------------------------------------------------------------------
extern "C" void kernel_launch(void* const* d_in, const int* in_sizes, int n_in,
                              void* d_out, int out_size, void* d_ws, size_t ws_size,
                              hipStream_t stream) {
    const int* rows = (const int*)d_in[0];
    const int* cols = (const int*)d_in[1];
    const float* adj = (const float*)d_in[2];
    const float* x = (const float*)d_in[3];
    const float* W = (const float*)d_in[4];
    const float* b = (const float*)d_in[5];
    float* out = (float*)d_out;

    const int n_edges = in_sizes[0];
    const int n_nodes = in_sizes[3] / IN_DIM;

    // Workspace layout: support [N*64] | deg [N] | dinv [N]
    float* support = (float*)d_ws;
    float* deg = support + (size_t)n_nodes * OUT_DIM;
    float* dinv = deg + n_nodes;

    const long long out_elems = (long long)n_nodes * OUT_DIM;

    // 1) zero out and deg (harness poisons buffers with 0xAA)
    {
        long long n = out_elems;
        zero_f32_kernel<<<(unsigned)((n + 255) / 256), 256, 0, stream>>>(out, n);
        zero_f32_kernel<<<(unsigned)((n_nodes + 255) / 256), 256, 0, stream>>>(deg, n_nodes);
    }

    // 2) degree scatter-add
    degree_kernel<<<(n_edges + 255) / 256, 256, 0, stream>>>(rows, adj, deg, n_edges);

    // 3) dinv = clamped deg^{-1/2}
    dinv_kernel<<<(n_nodes + 255) / 256, 256, 0, stream>>>(deg, dinv, n_nodes);

    // 4) support = x @ W + b  (WMMA f32 16x16x4; one wave per 16x64 block,
    //    2 waves per 64-thread block)
    {
        const int mtiles = (n_nodes + 15) / 16;         // 6250 for N=100000
        const int blocks = (mtiles + 1) / 2;            // 2 waves/block
        gcn_gemm_wmma<<<blocks, 64, 0, stream>>>(x, W, b, support, n_nodes);
    }

    // 5) SpMM scatter with hardware f32 atomics
    {
        const long long threads = (long long)n_edges * OUT_DIM;
        spmm_scatter_kernel<<<(unsigned)((threads + 255) / 256), 256, 0, stream>>>(
            rows, cols, adj, dinv, support, out, n_edges);
    }
}